// SkeletonLoss_71846212927821
// MI455X (gfx1250) — compile-verified
//
#include <hip/hip_runtime.h>
#include <math.h>

#define B_ 32
#define N_ 2048
#define NTILES (N_ / 16)

static constexpr float BIGF = 1e10f;
static constexpr float EPSF = 1e-12f;

typedef float v2f __attribute__((ext_vector_type(2)));
typedef float v8f __attribute__((ext_vector_type(8)));

// ---------------- workspace layout ----------------
// [0, B*N)   uint  : per-(b,m) column-min of d2 (float bits, values >= 0)
// [B*N, +B)  float : per-batch sum of sqrt(row_min)*pvf
// [.., +3)   float : accum[0]=sum point_sq, accum[1]=sum struct_sq, accum[2]=sum total_mask

__global__ __launch_bounds__(256) void init_ws_kernel(unsigned* colmin, float* sum_p,
                                                      float* accum) {
    int i = blockIdx.x * blockDim.x + threadIdx.x;
    if (i < B_ * N_) colmin[i] = __float_as_uint(BIGF);
    if (i < B_) sum_p[i] = 0.0f;
    if (i < 3) accum[i] = 0.0f;
}

// ---------------- point + structure losses (single pass over B*N) ----------------
__global__ __launch_bounds__(256) void point_struct_kernel(const float* __restrict__ pred,
                                                           const float* __restrict__ tgt,
                                                           const float* __restrict__ sk,
                                                           float* accum) {
    int idx = blockIdx.x * blockDim.x + threadIdx.x;
    float ps = 0.0f, ss = 0.0f, tm_sum = 0.0f;
    if (idx < B_ * N_) {
        float px = pred[idx * 3 + 0], py = pred[idx * 3 + 1];
        float tx = tgt[idx * 3 + 0], ty = tgt[idx * 3 + 1], tv = tgt[idx * 3 + 2];
        float vm = (tv == 1.0f) ? 1.0f : 0.0f;
        float dx = px - tx, dy = py - ty;
        float e2 = dx * dx + dy * dy;
        ps = e2 * vm;                            // (err*vm)^2 summed over 2 coords
        float sm = sk[idx * 2 + 0] + sk[idx * 2 + 1];
        sm = fminf(fmaxf(sm, 0.0f), 1.0f);
        float tm = sm * vm;
        ss = e2 * tm * tm;
        tm_sum = tm;
    }
    // wave32 reduction, one atomic per wave
    for (int m = 16; m >= 1; m >>= 1) {
        ps += __shfl_xor(ps, m);
        ss += __shfl_xor(ss, m);
        tm_sum += __shfl_xor(tm_sum, m);
    }
    if ((threadIdx.x & 31) == 0) {
        atomicAdd(&accum[0], ps);
        atomicAdd(&accum[1], ss);
        atomicAdd(&accum[2], tm_sum);
    }
}

// ---------------- chamfer core: WMMA f32 16x16x4 over d2 tiles ----------------
__global__ __launch_bounds__(256) void chamfer_kernel(const float* __restrict__ pred,
                                                      const float* __restrict__ tgt,
                                                      unsigned* __restrict__ gcolmin,
                                                      float* __restrict__ sum_p) {
    const int b  = blockIdx.x >> 4;   // 16 strip-blocks per batch (128 rows each)
    const int sb = blockIdx.x & 15;

    __shared__ float2   sTxy[N_];     // {tx, ty}
    __shared__ float2   sMB[N_];      // {|T|^2, tv?0:BIG}
    __shared__ unsigned sCol[N_];

    const int tid = threadIdx.x;
    for (int k = tid; k < N_; k += 256) {
        float tx = tgt[(b * N_ + k) * 3 + 0];
        float ty = tgt[(b * N_ + k) * 3 + 1];
        float tv = tgt[(b * N_ + k) * 3 + 2];
        sTxy[k] = make_float2(tx, ty);
        sMB[k]  = make_float2(tx * tx + ty * ty, (tv == 1.0f) ? 0.0f : BIGF);
        sCol[k] = __float_as_uint(BIGF);
    }
    __syncthreads();

    const int wave = tid >> 5;
    const int lane = tid & 31;
    const int half = lane >> 4;     // 0: lanes 0-15, 1: lanes 16-31
    const int nl   = lane & 15;
    const int rowbase = sb * 128 + wave * 16;

    // multiplicative K-pad mask: lanes 16-31 carry K=2,3 which must be zero.
    const float hm = (half == 0) ? 1.0f : 0.0f;

    // A operand (16x4 f32): lanes 0-15 hold K0/K1 (= -2x, -2y), lanes 16-31 hold K2/K3 = 0
    const int arow = rowbase + nl;
    float ax = pred[(b * N_ + arow) * 3 + 0];
    float ay = pred[(b * N_ + arow) * 3 + 1];
    v2f A;
    A.x = (-2.0f * ax) * hm;
    A.y = (-2.0f * ay) * hm;

    // per-lane row metadata for rows covered by this lane in the C/D tile:
    // lanes 0-15 hold rows rowbase+0..7 (VGPR i -> M=i), lanes 16-31 hold rows rowbase+8..15
    float pn2[8], pB[8], rmin[8];
#pragma unroll
    for (int i = 0; i < 8; ++i) {
        int r = rowbase + half * 8 + i;
        float px = pred[(b * N_ + r) * 3 + 0];
        float py = pred[(b * N_ + r) * 3 + 1];
        float pv = pred[(b * N_ + r) * 3 + 2];
        pn2[i]  = px * px + py * py;
        pB[i]   = (pv == 1.0f) ? 0.0f : BIGF;  // masked preds -> BIG via max()
        rmin[i] = BIGF;
    }

#pragma unroll 2
    for (int t = 0; t < NTILES; ++t) {
        const int cg = t * 16 + nl;            // global column (same for both halves)
        const float2 txy = sTxy[cg];           // one ds_load_b64
        const float2 mb  = sMB[cg];            // one ds_load_b64  (x=|T|^2, y=mask)

        // B operand (4x16 f32): lanes 0-15 hold K0/K1 rows, lanes 16-31 K2/K3 = 0
        v2f Bv;
        Bv.x = txy.x * hm;
        Bv.y = txy.y * hm;

        // preload C with |P|^2 + |T|^2 so the WMMA emits d2 directly
        v8f C;
#pragma unroll
        for (int i = 0; i < 8; ++i) C[i] = pn2[i] + mb.x;

        C = __builtin_amdgcn_wmma_f32_16x16x4_f32(
                /*neg_a=*/false, A, /*neg_b=*/false, Bv,
                /*c_mod=*/(short)0, C, /*reuse_a=*/false, /*reuse_b=*/false);

        float cpart = BIGF;
#pragma unroll
        for (int i = 0; i < 8; ++i) {
            float d2 = C[i];
            // max against {0,BIG}: applies visibility mask AND clamps tiny negative d2
            rmin[i] = fminf(rmin[i], fmaxf(d2, mb.y));
            cpart   = fminf(cpart, fmaxf(d2, pB[i]));
        }
        // combine halves: lane L has rows 0-7 min, lane L+16 has rows 8-15 min, same column
        cpart = fminf(cpart, __shfl_xor(cpart, 16));
        if (half == 0) atomicMin(&sCol[cg], __float_as_uint(cpart));
    }

    // row-min reduce across the 16 lanes of each half
#pragma unroll
    for (int i = 0; i < 8; ++i) {
#pragma unroll
        for (int m = 8; m >= 1; m >>= 1)
            rmin[i] = fminf(rmin[i], __shfl_xor(rmin[i], m));
    }
    if (nl == 0) {
        float s = 0.0f;
#pragma unroll
        for (int i = 0; i < 8; ++i) {
            float vis = (pB[i] == 0.0f) ? 1.0f : 0.0f;
            s += sqrtf(fmaxf(rmin[i], EPSF)) * vis;   // sqrt hoisted out of N^2 loop
        }
        atomicAdd(&sum_p[b], s);
    }

    __syncthreads();
    for (int k = tid; k < N_; k += 256)
        atomicMin(&gcolmin[b * N_ + k], sCol[k]);
}

// ---------------- finalize: per-batch chamfer + combine ----------------
__device__ __forceinline__ float block_sum_256(float v, float* red) {
    for (int m = 16; m >= 1; m >>= 1) v += __shfl_xor(v, m);
    const int wave = threadIdx.x >> 5, lane = threadIdx.x & 31;
    if (lane == 0) red[wave] = v;
    __syncthreads();
    float r = (threadIdx.x < 8) ? red[threadIdx.x] : 0.0f;
    if (wave == 0)
        for (int m = 4; m >= 1; m >>= 1) r += __shfl_xor(r, m);
    __syncthreads();
    if (threadIdx.x == 0) red[0] = r;
    __syncthreads();
    r = red[0];
    __syncthreads();
    return r;
}

__global__ __launch_bounds__(256) void finalize_kernel(const float* __restrict__ pred,
                                                       const float* __restrict__ tgt,
                                                       const unsigned* __restrict__ gcolmin,
                                                       const float* __restrict__ sum_p,
                                                       const float* __restrict__ accum,
                                                       float* __restrict__ out) {
    __shared__ float red[8];
    const int tid = threadIdx.x;
    float chamfer_sum = 0.0f;  // identical in every thread

    for (int b = 0; b < B_; ++b) {
        float cnt_p = 0.0f, cnt_t = 0.0f, sum_t = 0.0f;
        for (int k = tid; k < N_; k += 256) {
            float pv = pred[(b * N_ + k) * 3 + 2];
            float tv = tgt[(b * N_ + k) * 3 + 2];
            cnt_p += (pv == 1.0f) ? 1.0f : 0.0f;
            float tvis = (tv == 1.0f) ? 1.0f : 0.0f;
            cnt_t += tvis;
            float cm = __uint_as_float(gcolmin[b * N_ + k]);
            sum_t += sqrtf(fmaxf(cm, EPSF)) * tvis;
        }
        cnt_p = block_sum_256(cnt_p, red);
        cnt_t = block_sum_256(cnt_t, red);
        sum_t = block_sum_256(sum_t, red);
        float mean_p = sum_p[b] / fmaxf(cnt_p, 1.0f);
        float mean_t = sum_t / fmaxf(cnt_t, 1.0f);
        bool both = (cnt_p > 0.0f) && (cnt_t > 0.0f);
        chamfer_sum += both ? 0.5f * (mean_p + mean_t) : 0.0f;
    }

    if (tid == 0) {
        const float inv = 1.0f / (float)(B_ * N_ * 2);
        float lp = accum[0] * inv;
        float ls = (accum[2] > 0.0f) ? accum[1] * inv : 0.0f;
        float lc = chamfer_sum / (float)B_;
        out[0] = 1.0f * lp + 5.0f * lc + 2.0f * ls;  // W_POINT, W_CHAMFER, W_STRUCT
        out[1] = lp;
        out[2] = 0.0f;   // vis
        out[3] = lc;
    }
}

extern "C" void kernel_launch(void* const* d_in, const int* in_sizes, int n_in,
                              void* d_out, int out_size, void* d_ws, size_t ws_size,
                              hipStream_t stream) {
    const float* pred = (const float*)d_in[0];           // (B,N,3) f32
    const float* tgt  = (const float*)d_in[1];           // (B,N,3) f32
    const float* sk   = (const float*)d_in[2];           // (B,N,2) f32
    float* out = (float*)d_out;                          // 4 f32

    unsigned* colmin = (unsigned*)d_ws;
    float* sum_p = (float*)((char*)d_ws + (size_t)B_ * N_ * sizeof(unsigned));
    float* accum = sum_p + B_;

    init_ws_kernel<<<(B_ * N_ + 255) / 256, 256, 0, stream>>>(colmin, sum_p, accum);
    point_struct_kernel<<<(B_ * N_ + 255) / 256, 256, 0, stream>>>(pred, tgt, sk, accum);
    chamfer_kernel<<<B_ * 16, 256, 0, stream>>>(pred, tgt, colmin, sum_p);
    finalize_kernel<<<1, 256, 0, stream>>>(pred, tgt, colmin, sum_p, accum, out);
}